// GraphGather_47725676593209
// MI455X (gfx1250) — compile-verified
//
#include <hip/hip_runtime.h>

typedef __attribute__((ext_vector_type(2))) float v2f;
typedef __attribute__((ext_vector_type(8))) float v8f;

#define DCOL 256  // feature dim (reference: D=256), block = DCOL threads = 8 waves

__device__ __forceinline__ int lower_bound_ll(const long long* __restrict__ m,
                                              int n, long long v) {
  int lo = 0, hi = n;
  while (lo < hi) {
    int mid = (lo + hi) >> 1;
    if (m[mid] < v) lo = mid + 1; else hi = mid;
  }
  return lo;
}

// One block per segment. Wave w owns output columns [32w, 32w+32).
// Sum computed on the matrix pipe: C += A(16x4) x ones(4x16), features in A
// (documented 32-bit A layout: lane = M, K = 2*laneHalf + vgpr).
// Max computed on VALU, co-issuing with WMMA.
__global__ __launch_bounds__(DCOL) void seg_sum_max_wmma(
    const float* __restrict__ feat, const long long* __restrict__ mem,
    float* __restrict__ out, int n) {
  __shared__ int   s_bounds[2];
  __shared__ float s_sum[8 * 32];

  const int b   = blockIdx.x;
  const int tid = threadIdx.x;
  if (tid < 2) s_bounds[tid] = lower_bound_ll(mem, n, (long long)(b + tid));
  __syncthreads();
  const int start = s_bounds[0];
  const int end   = s_bounds[1];

  const int w  = tid >> 5;   // wave id 0..7
  const int l  = tid & 31;   // lane 0..31
  const int h  = l >> 4;     // lane half
  const int lm = l & 15;     // lane mod 16 (= M within a 16-col tile)
  const int cb = w * 32;     // this wave's column base

  v2f ones; ones.x = 1.0f; ones.y = 1.0f;   // B = ones(4x16): layout-invariant
  v8f c0 = {};  // accumulator, cols cb..cb+15
  v8f c1 = {};  // accumulator, cols cb+16..cb+31
  float vmax0 = 0.0f, vmax1 = 0.0f;         // 0-init is exact: output clamps max at 0

  // lane (l) loads A[m][k]: m = lm, k = 2h + vgpr  -> rows rb+2h, rb+2h+1
  const float* p = feat + (size_t)(start + 2 * h) * DCOL + cb + lm;

  int rb = start;
  #pragma unroll 2
  for (; rb + 3 < end; rb += 4) {
    float a00 = p[0];           // row rb+2h,   col cb+lm      (A0 vgpr0)
    float a01 = p[DCOL];        // row rb+2h+1, col cb+lm      (A0 vgpr1)
    float a10 = p[16];          // row rb+2h,   col cb+16+lm   (A1 vgpr0)
    float a11 = p[DCOL + 16];   // row rb+2h+1, col cb+16+lm   (A1 vgpr1)
    __builtin_prefetch(p + 16 * DCOL, 0, 0);  // global_prefetch_b8, ~4 chunks ahead
    p += 4 * DCOL;

    v2f A0; A0.x = a00; A0.y = a01;
    v2f A1; A1.x = a10; A1.y = a11;
    c0 = __builtin_amdgcn_wmma_f32_16x16x4_f32(false, A0, false, ones,
                                               (short)0, c0, false, false);
    c1 = __builtin_amdgcn_wmma_f32_16x16x4_f32(false, A1, false, ones,
                                               (short)0, c1, false, false);
    vmax0 = fmaxf(vmax0, fmaxf(a00, a01));
    vmax1 = fmaxf(vmax1, fmaxf(a10, a11));
  }

  if (rb < end) {  // tail chunk: pad missing rows with 0.0 (exact for sum and clamped max)
    const int r0 = rb + 2 * h;
    const int r1 = r0 + 1;
    const int r0c = (r0 < end) ? r0 : (end - 1);  // keep loads in-bounds
    const int r1c = (r1 < end) ? r1 : (end - 1);
    const float* q0 = feat + (size_t)r0c * DCOL + cb + lm;
    const float* q1 = feat + (size_t)r1c * DCOL + cb + lm;
    float a00 = (r0 < end) ? q0[0]  : 0.0f;
    float a01 = (r1 < end) ? q1[0]  : 0.0f;
    float a10 = (r0 < end) ? q0[16] : 0.0f;
    float a11 = (r1 < end) ? q1[16] : 0.0f;

    v2f A0; A0.x = a00; A0.y = a01;
    v2f A1; A1.x = a10; A1.y = a11;
    c0 = __builtin_amdgcn_wmma_f32_16x16x4_f32(false, A0, false, ones,
                                               (short)0, c0, false, false);
    c1 = __builtin_amdgcn_wmma_f32_16x16x4_f32(false, A1, false, ones,
                                               (short)0, c1, false, false);
    vmax0 = fmaxf(vmax0, fmaxf(a00, a01));
    vmax1 = fmaxf(vmax1, fmaxf(a10, a11));
  }

  // Extract column sums from D (all N identical since B is ones):
  // D layout: VGPR q, lanes 0-15 -> M=q ; lanes 16-31 -> M=q+8.
  // Lane 0 holds cols 0..7 of each tile (via q), lane 16 holds cols 8..15.
  if (lm == 0) {
    #pragma unroll
    for (int q = 0; q < 8; ++q) {
      s_sum[w * 32 + 8 * h + q]      = c0[q];
      s_sum[w * 32 + 16 + 8 * h + q] = c1[q];
    }
  }
  __syncthreads();

  const float sumv = s_sum[w * 32 + l];

  // combine max across lane halves (rows 2h,2h+1 mod 4 live in opposite halves)
  float m0 = fmaxf(vmax0, __shfl_xor(vmax0, 16, 32));
  float m1 = fmaxf(vmax1, __shfl_xor(vmax1, 16, 32));
  const float maxv = (h == 0) ? m0 : m1;  // lane l -> col cb + l in both cases

  const size_t obase = (size_t)b * (2 * DCOL);
  out[obase + cb + l]        = tanhf(sumv);
  out[obase + DCOL + cb + l] = tanhf(maxv);  // maxv >= 0 already (0-init)
}

extern "C" void kernel_launch(void* const* d_in, const int* in_sizes, int n_in,
                              void* d_out, int out_size, void* d_ws, size_t ws_size,
                              hipStream_t stream) {
  (void)n_in; (void)d_ws; (void)ws_size;
  const float*     feat = (const float*)d_in[0];
  const long long* mem  = (const long long*)d_in[1];  // membership is int64
  float*           out  = (float*)d_out;
  const int n = in_sizes[1];              // N atoms
  const int B = out_size / (2 * DCOL);    // segments
  seg_sum_max_wmma<<<dim3(B), dim3(DCOL), 0, stream>>>(feat, mem, out, n);
}